// HAN_58342835749545
// MI455X (gfx1250) — compile-verified
//
#include <hip/hip_runtime.h>
#include <hip/hip_bf16.h>
#include <math.h>

#define NN 50000
#define EE 800000
#define HH 8
#define FDIM 128
#define NEG_SLOPE 0.2f

typedef __attribute__((ext_vector_type(2))) float v2f;
typedef __attribute__((ext_vector_type(8))) float v8f;

// ---- helpers -------------------------------------------------------------

__device__ __forceinline__ void atomAddF(float* p, float v) {
    __hip_atomic_fetch_add(p, v, __ATOMIC_RELAXED, __HIP_MEMORY_SCOPE_AGENT);
}
__device__ __forceinline__ void atomMaxU(unsigned* p, unsigned v) {
    __hip_atomic_fetch_max(p, v, __ATOMIC_RELAXED, __HIP_MEMORY_SCOPE_AGENT);
}
// monotonic float<->uint mapping so unsigned atomic max == float max
__device__ __forceinline__ unsigned encodeF(float f) {
    unsigned u = __float_as_uint(f);
    return (u & 0x80000000u) ? ~u : (u | 0x80000000u);
}
__device__ __forceinline__ float decodeF(unsigned u) {
    return (u & 0x80000000u) ? __uint_as_float(u & 0x7fffffffu)
                             : __uint_as_float(~u);
}

// ---- 0: zero workspace ---------------------------------------------------

__global__ void zero_ws(unsigned* __restrict__ p, long long n) {
    long long i = (long long)blockIdx.x * blockDim.x + threadIdx.x;
    long long stride = (long long)gridDim.x * blockDim.x;
    for (; i < n; i += stride) p[i] = 0u;
}

// ---- 1: projection GEMM, WMMA f32 16x16x4 --------------------------------
// xp[m, n] = sum_k x[m,k] * w[k,n] + b[n]; one wave per 16x16 tile,
// 8 waves/block cover one 16-row strip of all 128 columns.

__global__ __launch_bounds__(256) void proj_wmma(const float* __restrict__ x,
                                                 const float* __restrict__ w,
                                                 const float* __restrict__ b,
                                                 float* __restrict__ xp) {
    const int lane = threadIdx.x & 31;
    const int wave = threadIdx.x >> 5;
    const int m0   = blockIdx.x << 4;
    const int n0   = wave << 4;
    const int half = lane >> 4;       // 0: lanes 0-15, 1: lanes 16-31
    const int r    = lane & 15;

    v8f acc = {0.f, 0.f, 0.f, 0.f, 0.f, 0.f, 0.f, 0.f};
    const float* arow = x + (size_t)(m0 + r) * FDIM + 2 * half; // A: M=r, K=2*half+{0,1}
    const float* bcol = w + (size_t)(2 * half) * FDIM + n0 + r; // B: N=r, K=2*half+{0,1}
    #pragma unroll 4
    for (int k0 = 0; k0 < FDIM; k0 += 4) {
        v2f a, bb;
        a.x  = arow[k0];
        a.y  = arow[k0 + 1];
        bb.x = bcol[(size_t)k0 * FDIM];
        bb.y = bcol[(size_t)(k0 + 1) * FDIM];
        acc = __builtin_amdgcn_wmma_f32_16x16x4_f32(false, a, false, bb,
                                                    (short)0, acc, false, false);
    }
    const float bias = b[n0 + r];
    float* orow = xp + (size_t)(m0 + 8 * half) * FDIM + n0 + r; // C: M=i+8*half, N=r
    #pragma unroll
    for (int i = 0; i < 8; ++i) orow[(size_t)i * FDIM] = acc[i] + bias;
}

// ---- 2: per-(node,head) attention coefficients ---------------------------

__global__ void attn_coef(const float* __restrict__ xp,
                          const float* __restrict__ ls0, const float* __restrict__ ld0,
                          const float* __restrict__ ls1, const float* __restrict__ ld1,
                          float* __restrict__ as0, float* __restrict__ ad0,
                          float* __restrict__ as1, float* __restrict__ ad1) {
    int i = blockIdx.x * blockDim.x + threadIdx.x;
    if (i >= NN * HH) return;
    const int h = i & 7;
    const float* xr = xp + (size_t)(i >> 3) * FDIM + h * 16;
    float s0 = 0.f, d0 = 0.f, s1 = 0.f, d1 = 0.f;
    #pragma unroll
    for (int d = 0; d < 16; ++d) {
        const float v = xr[d];
        s0 += v * ls0[h * 16 + d];  d0 += v * ld0[h * 16 + d];
        s1 += v * ls1[h * 16 + d];  d1 += v * ld1[h * 16 + d];
    }
    as0[i] = s0; ad0[i] = d0; as1[i] = s1; ad1[i] = d1;
}

// ---- 3: edge pass 1 — segment max of leaky_relu(alpha) per (dst, head) ---

__global__ void edge_max(const int* __restrict__ ei,
                         const float* __restrict__ as, const float* __restrict__ ad,
                         unsigned* __restrict__ mx) {
    int i = blockIdx.x * blockDim.x + threadIdx.x;
    if (i >= EE * HH) return;
    const int e = i >> 3, h = i & 7;
    const int s = ei[e], d = ei[EE + e];
    float al = as[s * 8 + h] + ad[d * 8 + h];
    al = al > 0.f ? al : NEG_SLOPE * al;
    atomMaxU(&mx[d * 8 + h], encodeF(al));
}

// ---- 4: edge pass 2 — segment sum of exp(alpha - max) --------------------

__global__ void edge_sum(const int* __restrict__ ei,
                         const float* __restrict__ as, const float* __restrict__ ad,
                         const unsigned* __restrict__ mx, float* __restrict__ den) {
    int i = blockIdx.x * blockDim.x + threadIdx.x;
    if (i >= EE * HH) return;
    const int e = i >> 3, h = i & 7;
    const int s = ei[e], d = ei[EE + e];
    float al = as[s * 8 + h] + ad[d * 8 + h];
    al = al > 0.f ? al : NEG_SLOPE * al;
    const float m = decodeF(mx[d * 8 + h]);
    atomAddF(&den[d * 8 + h], expf(al - m));
}

// ---- 5: edge pass 3 — weighted scatter-add of messages (wave per edge) ---

__global__ __launch_bounds__(256) void edge_aggr(const int* __restrict__ ei,
                                                 const float* __restrict__ as,
                                                 const float* __restrict__ ad,
                                                 const unsigned* __restrict__ mx,
                                                 const float* __restrict__ den,
                                                 const float* __restrict__ xp,
                                                 float* __restrict__ out) {
    const int e = blockIdx.x * (blockDim.x >> 5) + (threadIdx.x >> 5);
    if (e >= EE) return;
    const int lane = threadIdx.x & 31;
    const int s = ei[e], d = ei[EE + e];
    const float* xr = xp + (size_t)s * FDIM;
    float* orow = out + (size_t)d * FDIM;
    #pragma unroll
    for (int c = 0; c < 4; ++c) {
        const int f = lane + (c << 5);
        const int h = f >> 4;
        float al = as[s * 8 + h] + ad[d * 8 + h];
        al = al > 0.f ? al : NEG_SLOPE * al;
        const float m   = decodeF(mx[d * 8 + h]);
        const float wgt = expf(al - m) / (den[d * 8 + h] + 1e-16f);
        atomAddF(&orow[f], xr[f] * wgt);
    }
}

// ---- 6: k-GEMM (WMMA) with fused relu-in, tanh, per-column sum -----------
// ksum[n] += sum_rows tanh( relu(in) @ kw + kb )[.., n]

__global__ __launch_bounds__(256) void kgemm_wmma(const float* __restrict__ in,
                                                  const float* __restrict__ kw,
                                                  const float* __restrict__ kb,
                                                  float* __restrict__ ksum) {
    const int lane = threadIdx.x & 31;
    const int wave = threadIdx.x >> 5;
    const int m0   = blockIdx.x << 4;
    const int n0   = wave << 4;
    const int half = lane >> 4;
    const int r    = lane & 15;

    v8f acc = {0.f, 0.f, 0.f, 0.f, 0.f, 0.f, 0.f, 0.f};
    const float* arow = in + (size_t)(m0 + r) * FDIM + 2 * half;
    const float* bcol = kw + (size_t)(2 * half) * FDIM + n0 + r;
    #pragma unroll 4
    for (int k0 = 0; k0 < FDIM; k0 += 4) {
        v2f a, bb;
        a.x  = fmaxf(arow[k0], 0.f);          // relu fused on load
        a.y  = fmaxf(arow[k0 + 1], 0.f);
        bb.x = bcol[(size_t)k0 * FDIM];
        bb.y = bcol[(size_t)(k0 + 1) * FDIM];
        acc = __builtin_amdgcn_wmma_f32_16x16x4_f32(false, a, false, bb,
                                                    (short)0, acc, false, false);
    }
    const float bias = kb[n0 + r];
    float s = 0.f;
    #pragma unroll
    for (int i = 0; i < 8; ++i) s += tanhf(acc[i] + bias);
    atomAddF(&ksum[n0 + r], s);               // both half-waves add (16 rows total)
}

// ---- 7: semantic softmax over M=2 metapaths ------------------------------

__global__ void semantic(const float* __restrict__ ksum, const float* __restrict__ q,
                         float* __restrict__ attn) {
    if (threadIdx.x == 0 && blockIdx.x == 0) {
        float sc[2];
        for (int m = 0; m < 2; ++m) {
            float acc = 0.f;
            for (int f = 0; f < FDIM; ++f)
                acc += q[f] * ksum[m * FDIM + f] * (1.0f / (float)NN);
            sc[m] = acc;
        }
        const float mx = fmaxf(sc[0], sc[1]);
        const float e0 = expf(sc[0] - mx), e1 = expf(sc[1] - mx);
        const float inv = 1.f / (e0 + e1);
        attn[0] = e0 * inv;
        attn[1] = e1 * inv;
    }
}

// ---- 8: final combine (relu fused) ---------------------------------------

__global__ void combine(const float* __restrict__ o0, const float* __restrict__ o1,
                        const float* __restrict__ attn, float* __restrict__ out) {
    int i = blockIdx.x * blockDim.x + threadIdx.x;
    if (i >= NN * FDIM) return;
    out[i] = attn[0] * fmaxf(o0[i], 0.f) + attn[1] * fmaxf(o1[i], 0.f);
}

// ---- launch --------------------------------------------------------------

extern "C" void kernel_launch(void* const* d_in, const int* in_sizes, int n_in,
                              void* d_out, int out_size, void* d_ws, size_t ws_size,
                              hipStream_t stream) {
    const float* x    = (const float*)d_in[0];
    const int*   ei0  = (const int*)d_in[1];
    const int*   ei1  = (const int*)d_in[2];
    const float* pw   = (const float*)d_in[3];
    const float* pb   = (const float*)d_in[4];
    const float* ls0  = (const float*)d_in[5];
    const float* ld0  = (const float*)d_in[6];
    const float* ls1  = (const float*)d_in[7];
    const float* ld1  = (const float*)d_in[8];
    const float* kw   = (const float*)d_in[9];
    const float* kb   = (const float*)d_in[10];
    const float* q    = (const float*)d_in[11];
    float* out = (float*)d_out;

    // workspace layout (floats unless noted)
    float*    xp    = (float*)d_ws;                        // N*128
    float*    out0  = xp   + (size_t)NN * FDIM;            // N*128
    float*    out1  = out0 + (size_t)NN * FDIM;            // N*128
    float*    as0   = out1 + (size_t)NN * FDIM;            // N*H
    float*    ad0   = as0  + (size_t)NN * HH;
    float*    as1   = ad0  + (size_t)NN * HH;
    float*    ad1   = as1  + (size_t)NN * HH;
    unsigned* mx0   = (unsigned*)(ad1 + (size_t)NN * HH);  // N*H u32
    unsigned* mx1   = mx0 + (size_t)NN * HH;
    float*    den0  = (float*)(mx1 + (size_t)NN * HH);     // N*H
    float*    den1  = den0 + (size_t)NN * HH;
    float*    ksum  = den1 + (size_t)NN * HH;              // 2*128
    float*    attn  = ksum + 2 * FDIM;                     // 2

    // zero out0..attn (atomically-accumulated buffers); xp is fully written
    const long long zcount = 2LL * NN * FDIM + 8LL * NN * HH + 2 * FDIM + 2;
    zero_ws<<<2048, 256, 0, stream>>>((unsigned*)out0, zcount);

    // projection GEMM (WMMA): 3125 row-strips x (8 waves = all 128 cols)
    proj_wmma<<<NN / 16, 256, 0, stream>>>(x, pw, pb, xp);

    // per-(node,head) attention coefficients for both metapaths
    attn_coef<<<(NN * HH + 255) / 256, 256, 0, stream>>>(xp, ls0, ld0, ls1, ld1,
                                                         as0, ad0, as1, ad1);

    const int egrid = (EE * HH + 255) / 256;       // thread per (edge, head)
    const int agrid = (EE + 7) / 8;                // wave per edge, 8 waves/block

    // metapath 0
    edge_max <<<egrid, 256, 0, stream>>>(ei0, as0, ad0, mx0);
    edge_sum <<<egrid, 256, 0, stream>>>(ei0, as0, ad0, mx0, den0);
    edge_aggr<<<agrid, 256, 0, stream>>>(ei0, as0, ad0, mx0, den0, xp, out0);

    // metapath 1
    edge_max <<<egrid, 256, 0, stream>>>(ei1, as1, ad1, mx1);
    edge_sum <<<egrid, 256, 0, stream>>>(ei1, as1, ad1, mx1, den1);
    edge_aggr<<<agrid, 256, 0, stream>>>(ei1, as1, ad1, mx1, den1, xp, out1);

    // semantic attention: k = tanh(relu(out_m) @ kw + kb).mean(rows)
    kgemm_wmma<<<NN / 16, 256, 0, stream>>>(out0, kw, kb, ksum);
    kgemm_wmma<<<NN / 16, 256, 0, stream>>>(out1, kw, kb, ksum + FDIM);
    semantic<<<1, 32, 0, stream>>>(ksum, q, attn);

    // weighted combine (relu fused)
    combine<<<(NN * FDIM + 255) / 256, 256, 0, stream>>>(out0, out1, attn, out);
}